// EPGSimulationMQC_44435731644532
// MI455X (gfx1250) — compile-verified
//
#include <hip/hip_runtime.h>

typedef __attribute__((ext_vector_type(2))) float v2f;
typedef __attribute__((ext_vector_type(8))) float v8f;

namespace {
constexpr int kNStates      = 4096;
constexpr int kNPulses      = 512;
constexpr int kSStr         = 10;    // floats per state row: Fm2,Fm1,Fp1,Fp2 (re,im) + Z (re,im)
constexpr int kThreads      = 1024;  // 32 wave32 waves -> one WGP-resident workgroup
constexpr int kTilesPerWave = 8;     // 4096/16 tiles / 32 waves
constexpr int kSLen         = kNStates * kSStr;  // 40960 floats
constexpr int kDummyLds     = kSLen;             // 128-float dummy tail for spectator lanes
constexpr unsigned kFsBytesPerT = (unsigned)kNStates * kSStr * 4;  // 163840
constexpr unsigned kZsBase      = (unsigned)kNPulses * kNStates * kSStr * 4;  // byte offset of Zs
}

// Per-pulse 12x16 real update matrix (row-major MB[n][k], stride 12). Folds E2 decay,
// B0 precession, RF mixing (incl. conjugations) and E1 relaxation/regrowth (homogeneous
// k=10 column) into one linear map y = M x with
//   x = [Fm2r,Fm2i, Fm1r,Fm1i, Fp1r,Fp1i, Fp2r,Fp2i, Zr,Zi, 1, 0]
//   y = [nm2r,nm2i, nm1r,nm1i, np1r,np1i, np2r,np2i, Znr,Zni, 0...]
__device__ __forceinline__ void build_step_matrix(
    float* __restrict__ MB, float fa, float ph, float B1v,
    float E1, float E2, float rE1,
    float phi, float cphi, float sphi, float c2phi, float s2phi) {
  float ss, cc;
  sincosf(0.5f * fa * B1v, &ss, &cc);
  const float c2 = cc * cc, s2 = ss * ss, cs = cc * ss, k2 = c2 - s2;
  float sph, cph;  sincosf(ph, &sph, &cph);
  float sb1, cb1;  sincosf(phi + 2.0f * ph, &sb1, &cb1);
  float sb2, cb2;  sincosf(2.0f * phi + 4.0f * ph, &sb2, &cb2);
  float sg,  cg;   sincosf(ph - phi, &sg, &cg);
  const float A1r = c2 * E2 * cphi,  A1i = c2 * E2 * sphi;   // c^2 E2 e^{i phi}
  const float B1r = s2 * E2 * cb1,   B1i = s2 * E2 * sb1;    // s^2 E2 e^{i(phi+2ph)}
  const float A2r = c2 * E2 * c2phi, A2i = c2 * E2 * s2phi;  // c^2 E2 e^{2i phi}
  const float B2r = s2 * E2 * cb2,   B2i = s2 * E2 * sb2;    // s^2 E2 e^{i(2phi+4ph)}
  const float C1r = -cs * E1 * sph,  C1i = cs * E1 * cph;    // i cs E1 e^{i ph}
  const float K1r = -cs * rE1 * sph, K1i = cs * rE1 * cph;   // i cs (1-E1) e^{i ph}
  const float Gr  = cs * E2 * sg,    Gi  = -cs * E2 * cg;    // -i cs E2 e^{i(ph-phi)}
  const float ZC  = k2 * E1,         ZK  = k2 * rE1;
  float* r;
  r = MB + 0*12;  r[0]= A2r; r[1]= A2i; r[6]= B2r; r[7]=-B2i;                                   // nm2r
  r = MB + 1*12;  r[0]=-A2i; r[1]= A2r; r[6]=-B2i; r[7]=-B2r;                                   // nm2i
  r = MB + 2*12;  r[2]= A1r; r[3]= A1i; r[4]= B1r; r[5]=-B1i; r[8]= C1r; r[9]= C1i; r[10]= K1r; // nm1r
  r = MB + 3*12;  r[2]=-A1i; r[3]= A1r; r[4]=-B1i; r[5]=-B1r; r[8]=-C1i; r[9]= C1r; r[10]=-K1i; // nm1i
  r = MB + 4*12;  r[2]= B1r; r[3]= B1i; r[4]= A1r; r[5]=-A1i; r[8]= C1r; r[9]=-C1i; r[10]= K1r; // np1r
  r = MB + 5*12;  r[2]= B1i; r[3]=-B1r; r[4]= A1i; r[5]= A1r; r[8]= C1i; r[9]= C1r; r[10]= K1i; // np1i
  r = MB + 6*12;  r[0]= B2r; r[1]= B2i; r[6]= A2r; r[7]=-A2i;                                   // np2r
  r = MB + 7*12;  r[0]= B2i; r[1]=-B2r; r[6]= A2i; r[7]= A2r;                                   // np2i
  r = MB + 8*12;  r[2]= Gr;  r[3]=-Gi;  r[4]= Gr;  r[5]= Gi;  r[8]= ZC;  r[10]= ZK;             // Znr
  r = MB + 9*12;  r[2]= Gi;  r[3]= Gr;  r[4]=-Gi;  r[5]= Gr;  r[9]= ZC;                         // Zni
}

__launch_bounds__(kThreads, 1)
__global__ void epg_mqc_wmma_kernel(const float* __restrict__ flip,
                                    const float* __restrict__ phase,
                                    const float* __restrict__ pT1,
                                    const float* __restrict__ pT2,
                                    const float* __restrict__ pTR,
                                    const float* __restrict__ pB0,
                                    const float* __restrict__ pB1,
                                    float* __restrict__ out) {
  __shared__ float S[kSLen + 128];  // 160 KB EPG state + dummy tail (CDNA5 WGP: 320 KB LDS)
  __shared__ float MB[2][16 * 12];  // double-buffered per-pulse matrix (rows 10..15 stay 0)

  const int tid  = threadIdx.x;
  const int wid  = tid >> 5;
  const int lane = tid & 31;
  const int h    = lane >> 4;  // fragment half
  const int n    = lane & 15;  // A state row / B column / D feature index
  const int base = wid * (kTilesPerWave * 16);

  const float T1 = pT1[0], T2 = pT2[0], TR = pTR[0];
  const float B0 = pB0[0], B1v = pB1[0];
  const float E1  = __expf(-TR / T1);
  const float E2  = __expf(-TR / T2);
  const float rE1 = 1.0f - E1;
  const float phi = 6.28318530717958647f * B0 * TR * 0.001f;
  float sphi, cphi;   sincosf(phi, &sphi, &cphi);
  float s2phi, c2phi; sincosf(2.0f * phi, &s2phi, &c2phi);

  // ---------------- fixed per-lane roles (branch-free writeback) ----------------
  const bool isF    = (n <= 7);             // shifted F feature -> Fs + S[st+1]
  const bool isZ    = (n == 8 || n == 9);   // Z feature -> Zs + S[st]
  const bool isZero = (n == 10 || n == 11); // D col exactly 0 -> Fs q=0 column
  const bool fType  = isF || isZero;
  const int  pos    = (n < 4) ? n : (n <= 7 ? n + 2 : n - 6);  // float pos in Fs row
  // global stores: byte offset off(t, st) = gOffB + st*gStrideB; gOffB += gTStrideB per t.
  // Spectator lanes (n>=12, value always exactly 0) alias unique always-zero q=0 cells.
  unsigned gOffB, gStrideB, gTStrideB;
  if (fType)    { gOffB = (unsigned)(kSStr + pos) * 4;            gStrideB = 40; gTStrideB = kFsBytesPerT; }
  else if (isZ) { gOffB = kZsBase + (unsigned)(n - 8) * 4;        gStrideB = 8;  gTStrideB = (unsigned)kNStates * 8; }
  else          { gOffB = ((unsigned)(wid * 2 + ((n - 12) >> 1)) * kSStr + 4 + (n & 1)) * 4;
                  gStrideB = 0; gTStrideB = 0; }
  const bool fRedir = (h == 1) && fType;    // lanes whose r=7 row can be st=4095
  // LDS writeback: idx(st) = ldsOff + st*ldsMul, then +10 floats per row (imm offsets)
  unsigned ldsOff, ldsMul, ldsOff7, ldsMul7, pendOff, pendMul;
  if (isF)      { ldsOff = kSStr + n; ldsMul = kSStr; }          // S[(st+1)*10 + n]
  else if (isZ) { ldsOff = n;         ldsMul = kSStr; }          // S[st*10 + n]
  else          { ldsOff = kDummyLds + lane; ldsMul = 0; }
  if (isF && h == 1) { ldsOff7 = kDummyLds + lane; ldsMul7 = 0; }  // deferred via pend
  else               { ldsOff7 = ldsOff;           ldsMul7 = ldsMul; }
  if (isF && h == 1) { pendOff = n;                pendMul = kSStr; }  // S[tile0*10 + n]
  else               { pendOff = kDummyLds + lane; pendMul = 0; }
  const int stp = base + kTilesPerWave * 16;  // cross-wave shift target
  const unsigned pendXIdx = (isF && h == 1 && stp < kNStates)
                              ? (unsigned)(stp * kSStr + n) : (unsigned)(kDummyLds + lane);

  // ---------------- init: F = 0, Z[0] = 1, matrix buffers zeroed ----------------
  for (int i = tid; i < kSLen + 128; i += kThreads) S[i] = 0.0f;
  float* mbflat = &MB[0][0];
  for (int i = tid; i < 2 * 16 * 12; i += kThreads) mbflat[i] = 0.0f;
  __syncthreads();
  if (tid == 0) {
    S[8] = 1.0f;  // Z[0] = 1 + 0i
    build_step_matrix(&MB[0][0], flip[0], phase[0], B1v, E1, E2, rE1,
                      phi, cphi, sphi, c2phi, s2phi);
  }
  __syncthreads();

  #pragma unroll 1
  for (int t = 0; t < kNPulses; ++t) {
    // B fragments for V_WMMA_F32_16X16X4_F32: pair = B[k=4c+2h][n], B[k=4c+2h+1][n]
    const float* brow = &MB[t & 1][0] + n * 12 + 2 * h;
    const v2f b0 = *(const v2f*)(brow + 0);
    const v2f b1 = *(const v2f*)(brow + 4);
    const v2f b2 = *(const v2f*)(brow + 8);

    float pend = 0.0f;
    #pragma unroll 1
    for (int j = 0; j < kTilesPerWave; ++j) {
      const int tile0 = base + j * 16;
      // ---- A fragments (branch-free): lane holds features (4c+2h, 4c+2h+1) ----
      const float* row = &S[(tile0 + n) * kSStr + 2 * h];
      const v2f a0 = *(const v2f*)(row);        // h0: Fm2, h1: Fm1
      const v2f a1 = *(const v2f*)(row + 4);    // h0: Fp1, h1: Fp2
      const v2f t2 = *(const v2f*)(&S[(tile0 + n) * kSStr + 8]);
      v2f a2;                                   // h0: Z, h1: homogeneous (1,0)
      a2.x = h ? 1.0f : t2.x;
      a2.y = h ? 0.0f : t2.y;
      // deferred shift value from previous tile (row tile0 already read above)
      if (j > 0) S[pendOff + (unsigned)tile0 * pendMul] = pend;

      // ---- D(16 states x 16 feats) = A x B, K=12 via 3 chained f32 WMMAs ----
      v8f acc = {};
      acc = __builtin_amdgcn_wmma_f32_16x16x4_f32(false, a0, false, b0,
                                                  (short)0, acc, false, false);
      acc = __builtin_amdgcn_wmma_f32_16x16x4_f32(false, a1, false, b1,
                                                  (short)0, acc, false, false);
      acc = __builtin_amdgcn_wmma_f32_16x16x4_f32(false, a2, false, b2,
                                                  (short)0, acc, false, false);

      // ---- writeback: VGPR r holds feature n of state st = tile0 + r + 8h ----
      const unsigned stBase = (unsigned)(tile0 + 8 * h);
      const unsigned ldsA   = ldsOff + stBase * ldsMul;   // +10 floats per row (imm)
      unsigned       o      = gOffB + stBase * gStrideB;  // byte offset vs uniform base
      #pragma unroll
      for (int r = 0; r < 7; ++r) {
        const float v = acc[r];
        S[ldsA + r * kSStr] = v;
        __builtin_nontemporal_store(v, (float*)((char*)out + o));
        o += gStrideB;
      }
      {  // r = 7: tile-boundary row (h1 F lanes defer via pend; wave31/j7 drops st=4095)
        const float v = acc[7];
        S[ldsOff7 + (stBase + 7u) * ldsMul7] = v;
        pend = v;
        if (wid == 31 && j == kTilesPerWave - 1) {
          if (!fRedir) __builtin_nontemporal_store(v, (float*)((char*)out + o));
        } else {
          __builtin_nontemporal_store(v, (float*)((char*)out + o));
        }
      }
    }

    // Fsh[0] = 0 every step
    if (wid == 0 && lane < 10)
      __builtin_nontemporal_store(
          0.0f, (float*)((char*)out + ((unsigned)t * kFsBytesPerT + (unsigned)lane * 4)));

    // overlap next pulse's matrix build with other waves finishing their tiles
    if (tid == 0 && t + 1 < kNPulses)
      build_step_matrix(&MB[(t + 1) & 1][0], flip[t + 1], phase[t + 1], B1v,
                        E1, E2, rE1, phi, cphi, sphi, c2phi, s2phi);

    __syncthreads();           // all state reads of step t complete
    S[pendXIdx] = pend;        // cross-wave shift write (branch-free via dummy slot)
    gOffB += gTStrideB;
    __syncthreads();           // state(t+1) fully built before next step's reads
  }
}

extern "C" void kernel_launch(void* const* d_in, const int* in_sizes, int n_in,
                              void* d_out, int out_size, void* d_ws, size_t ws_size,
                              hipStream_t stream) {
  (void)in_sizes; (void)n_in; (void)out_size; (void)d_ws; (void)ws_size;
  const float* flip  = (const float*)d_in[0];
  const float* phase = (const float*)d_in[1];
  const float* T1    = (const float*)d_in[2];
  const float* T2    = (const float*)d_in[3];
  const float* TR    = (const float*)d_in[4];
  // d_in[5] = TE, unused by the reference computation
  const float* B0    = (const float*)d_in[6];
  const float* B1    = (const float*)d_in[7];
  epg_mqc_wmma_kernel<<<dim3(1), dim3(kThreads), 0, stream>>>(
      flip, phase, T1, T2, TR, B0, B1, (float*)d_out);
}